// MWPT_90787018703391
// MI455X (gfx1250) — compile-verified
//
#include <hip/hip_runtime.h>

// CDNA5 / gfx1250, wave32. Fused 3-level wavelet packet transform.
// Convs on the matrix pipe via V_WMMA_F32_16X16X4_F32; input staged
// HBM->LDS via GLOBAL_LOAD_ASYNC_TO_LDS_B32 when the builtin exists.

typedef __attribute__((ext_vector_type(2))) float v2f;
typedef __attribute__((ext_vector_type(8))) float v8f;

#if __has_builtin(__builtin_amdgcn_global_load_async_to_lds_b32)
#define HAVE_ASYNC_LDS 1
typedef __attribute__((address_space(1))) int* as1_ip;   // global
typedef __attribute__((address_space(3))) int* as3_ip;   // LDS
#endif

#define T_LEN   65536
#define T_MASK  (T_LEN - 1)
#define N3      256            // level-3 samples per tile (per node)
#define L3W     256            // = 2 * 128
#define L2W     640            // = 5 * 128  (range [m0-64,  m0+576))
#define L1W     1408           // = 11 * 128 (range [l0-192, l0+1216))
#define IN_W    2832           // staged input floats (2822 used, padded)
#define L1_HALO 192
#define L2_HALO 64

// One 16x16 WMMA-tile job: 128 consecutive outputs for BOTH filters.
//   out_f[O0 + 8m + p] = sum_t src_logical[2*(O0+8m+p) + t - 3] * w_f[t]
// mapped as D[m, f*8+p] = sum_k A[m,k] * B[k, f*8+p],
//   A[m,k] = src[src_shift + 16m + k]  (src_shift = lds offset of 2*O0-3)
//   B[k,n] = w_f[k - 2p]  (0 outside 0..7), precomputed per-lane in bfrag.
__device__ __forceinline__ void conv_tile(const float* src, int src_shift,
                                          float* dst_lo, float* dst_hi,
                                          int dst_off, const v2f* bfrag, int lane)
{
    const int mrow  = lane & 15;            // A row (both lane halves hold M=0..15)
    const int khalf = (lane >> 4) << 1;     // lanes 16-31 carry K+2, K+3
    const int abase = src_shift + 16 * mrow + khalf;

    v8f acc = {0.f, 0.f, 0.f, 0.f, 0.f, 0.f, 0.f, 0.f};
#pragma unroll
    for (int kk = 0; kk < 6; ++kk) {        // K = 24 (taps 0..21 live, 22..23 zero-B)
        v2f a;
        a.x = src[abase + 4 * kk];
        a.y = src[abase + 4 * kk + 1];
        acc = __builtin_amdgcn_wmma_f32_16x16x4_f32(
            /*neg_a=*/false, a, /*neg_b=*/false, bfrag[kk],
            /*c_mod=*/(short)0, acc, /*reuse_a=*/false, /*reuse_b=*/false);
    }

    // D layout: VGPR v, lane l -> row m = v + 8*(l>>4), col n = l&15.
    // col n = f*8 + p  ->  out_f[O0 + 8m + p]
    float* dst = ((lane & 15) >> 3) ? dst_hi : dst_lo;
    const int base = dst_off + ((lane >> 4) << 6) + (lane & 7);
#pragma unroll
    for (int v = 0; v < 8; ++v)
        dst[base + 8 * v] = acc[v];
}

__global__ __launch_bounds__(256)
void mwpt_fused(const float* __restrict__ x, const float* __restrict__ ker,
                float* __restrict__ out)
{
    __shared__ __align__(16) float s_in[IN_W];
    __shared__ __align__(16) float s_l1[2][L1W];   // [a0, a4]
    __shared__ __align__(16) float s_l2[4][L2W];   // [b0, b2, b4, b6]
    __shared__ __align__(16) float s_l3[8][L3W];   // [c0..c7]
    __shared__ __align__(16) float s_w[16];        // [w_lo(8), w_hi(8)]

    const int tid  = threadIdx.x;
    const int lane = tid & 31;
    const int uw   = __builtin_amdgcn_readfirstlane(tid >> 5);  // uniform wave id
    const int row  = blockIdx.x >> 5;    // 32 tiles per row
    const int tile = blockIdx.x & 31;

    const int n0 = tile * N3;            // level-3 output base
    const int m0 = 2 * n0;               // level-2 output base
    const int l0 = 4 * n0;               // level-1 output base

    // Filters: w_lo = kernel, w_hi[t] = kernel[7-t] * (-1)^t  (QMF)
    if (tid < 8) {
        s_w[tid] = ker[tid];
    } else if (tid < 16) {
        int t = tid - 8;
        s_w[8 + t] = ker[7 - t] * ((t & 1) ? -1.0f : 1.0f);
    }

    // Stage input tile + halo; circular wrap applied exactly once, here.
    const float* xr = x + (size_t)row * T_LEN;
    const int in_base = 2 * l0 - 387;    // logical index of s_in[0]
    for (int i = tid; i < IN_W; i += 256) {
        int g = (in_base + i + T_LEN) & T_MASK;
#ifdef HAVE_ASYNC_LDS
        __builtin_amdgcn_global_load_async_to_lds_b32(
            (as1_ip)(xr + g), (as3_ip)&s_in[i], /*offset=*/0, /*cpol=*/0);
#else
        s_in[i] = xr[g];
#endif
    }
#ifdef HAVE_ASYNC_LDS
#if __has_builtin(__builtin_amdgcn_s_wait_asynccnt)
    __builtin_amdgcn_s_wait_asynccnt(0);
#else
    asm volatile("s_wait_asynccnt 0x0" ::: "memory");
#endif
#endif
    __syncthreads();

    // Per-lane constant B fragments (same matrix at every level/tile).
    v2f bfrag[6];
    {
        const int ncol  = lane & 15;
        const int fsel  = ncol >> 3;     // 0 = low-pass col, 1 = high-pass col
        const int p     = ncol & 7;      // output phase within col group
        const int khalf = (lane >> 4) << 1;
#pragma unroll
        for (int kk = 0; kk < 6; ++kk) {
            int i0 = 4 * kk + khalf - 2 * p;
            int i1 = i0 + 1;
            bfrag[kk].x = (i0 >= 0 && i0 < 8) ? s_w[fsel * 8 + i0] : 0.0f;
            bfrag[kk].y = (i1 >= 0 && i1 < 8) ? s_w[fsel * 8 + i1] : 0.0f;
        }
    }

    // Level 1: x -> (a0, a4); 11 tiles over [l0-192, l0+1216)
    for (int j = uw; j < 11; j += 8)
        conv_tile(s_in, 256 * j, s_l1[0], s_l1[1], 128 * j, bfrag, lane);
    __syncthreads();

    // Level 2: a0->(b0,b2), a4->(b4,b6); 5 tiles each over [m0-64, m0+576)
    for (int q = uw; q < 10; q += 8) {
        int sn = q / 5, tj = q % 5;
        conv_tile(s_l1[sn], 61 + 256 * tj,
                  s_l2[2 * sn], s_l2[2 * sn + 1], 128 * tj, bfrag, lane);
    }
    __syncthreads();

    // Level 3: b_{2i} -> (c_{2i}, c_{2i+1}); 2 tiles each over [n0, n0+256)
    for (int q = uw; q < 8; q += 8) {
        int sn = q >> 1, tj = q & 1;
        conv_tile(s_l2[sn], 61 + 256 * tj,
                  s_l3[2 * sn], s_l3[2 * sn + 1], 128 * tj, bfrag, lane);
    }
    __syncthreads();

    // Vectorized (float4 / b128) coalesced writeback.
    // Channel -> LDS node is the gray code c ^ (c>>1) at every level.
    float4* out4 = (float4*)out;
    for (int i = tid; i < 512; i += 256) {          // level1: [128,2,32768]
        int nd = i >> 8, l4 = i & 255;
        out4[(size_t)(row * 2 + nd) * 8192 + (l0 >> 2) + l4]
            = *(const float4*)&s_l1[nd][L1_HALO + 4 * l4];
    }
    for (int i = tid; i < 512; i += 256) {          // level2: [128,4,16384]
        int c = i >> 7, m4 = i & 127;
        out4[2097152u + (size_t)(row * 4 + c) * 4096 + (m0 >> 2) + m4]
            = *(const float4*)&s_l2[c ^ (c >> 1)][L2_HALO + 4 * m4];
    }
    for (int i = tid; i < 512; i += 256) {          // level3: [128,8,8192]
        int c = i >> 6, n4 = i & 63;
        out4[4194304u + (size_t)(row * 8 + c) * 2048 + (n0 >> 2) + n4]
            = *(const float4*)&s_l3[c ^ (c >> 1)][4 * n4];
    }
}

extern "C" void kernel_launch(void* const* d_in, const int* in_sizes, int n_in,
                              void* d_out, int out_size, void* d_ws, size_t ws_size,
                              hipStream_t stream) {
    const float* x   = (const float*)d_in[0];   // [128,1,65536] f32
    const float* ker = (const float*)d_in[1];   // [1,1,8] f32
    float* out = (float*)d_out;                 // level1|level2|level3 concat

    dim3 grid(128 * 32);   // 128 rows x 32 tiles of 256 level-3 samples
    dim3 block(256);       // 8 wave32 waves
    hipLaunchKernelGGL(mwpt_fused, grid, block, 0, stream, x, ker, out);
}